// CapLayer_68779606278970
// MI455X (gfx1250) — compile-verified
//
#include <hip/hip_runtime.h>

typedef __attribute__((ext_vector_type(2))) float v2f;
typedef __attribute__((ext_vector_type(8))) float v8f;

#define NUM_SHARED 32
#define IN_DIM     8
#define NUM_OUT    10
#define OUT_DIM    16
#define HW         36                  // 6*6
#define I_TOT      (NUM_SHARED * HW)   // 1152
#define ROUTE_NUM  3
#define EPS_SQ     1e-20f

__global__ __launch_bounds__(128) void caps_route_kernel(
    const float* __restrict__ x,      // (bs, 256, 6, 6)
    const float* __restrict__ w,      // (32, 160, 8)
    const float* __restrict__ bconv,  // (5120,)
    float* __restrict__ out)          // (bs, 10, 16)
{
    __shared__ float pred[OUT_DIM * I_TOT];  // 73728 B: pred[d][i]
    __shared__ float blog[I_TOT];            // routing logits b
    __shared__ float cprob[I_TOT];           // softmax c
    __shared__ float red[64];                // per-wave partials (4 waves x 16)
    __shared__ float sv[OUT_DIM];
    __shared__ float vv[OUT_DIM];

    const int tid  = threadIdx.x;
    const int wave = tid >> 5;
    const int lane = tid & 31;
    const int half = lane >> 4;      // 0: lanes 0-15, 1: lanes 16-31
    const int l    = lane & 15;

    const int j  = blockIdx.x % NUM_OUT;
    const int bb = blockIdx.x / NUM_OUT;

    // ---------------- Phase 1: pred[d][i] via V_WMMA_F32_16X16X4_F32 --------
    // Per group s: D(16x36) = W_s(16x8) * X_s(8x36) + bias, tiled N into 3x16.
    const float* xb = x + (size_t)bb * (NUM_SHARED * IN_DIM * HW);
    for (int s = wave * 8; s < wave * 8 + 8; ++s) {
        // A fragments (16x4 f32, 2 VGPRs): lane m = l; VGPR0 holds K=2*half,
        // VGPR1 holds K=2*half+1; second k-step adds 4.
        const float* wrow = w + (size_t)(s * 160 + j * 16 + l) * 8;
        v2f a0, a1;
        a0[0] = wrow[2 * half + 0];
        a0[1] = wrow[2 * half + 1];
        a1[0] = wrow[2 * half + 4];
        a1[1] = wrow[2 * half + 5];
        // bias for accumulator rows r+8*half
        float cb[8];
#pragma unroll
        for (int r = 0; r < 8; ++r)
            cb[r] = bconv[s * 160 + j * 16 + r + 8 * half];

        const float* xs = xb + (size_t)(s * IN_DIM) * HW;
        for (int nt = 0; nt < 3; ++nt) {
            const int  p  = nt * 16 + l;
            const bool pv = (p < HW);
            // B fragments (4x16 f32, 2 VGPRs): col n = l; VGPR0 row K=2*half,
            // VGPR1 row K=2*half+1; second k-step adds 4. Tail cols zeroed.
            v2f b0, b1;
            const int k0 = 2 * half;
            b0[0] = pv ? xs[(k0 + 0) * HW + p] : 0.0f;
            b0[1] = pv ? xs[(k0 + 1) * HW + p] : 0.0f;
            b1[0] = pv ? xs[(k0 + 4) * HW + p] : 0.0f;
            b1[1] = pv ? xs[(k0 + 5) * HW + p] : 0.0f;

            v8f c;
#pragma unroll
            for (int r = 0; r < 8; ++r) c[r] = cb[r];

            c = __builtin_amdgcn_wmma_f32_16x16x4_f32(
                    false, a0, false, b0, (short)0, c, false, false);
            c = __builtin_amdgcn_wmma_f32_16x16x4_f32(
                    false, a1, false, b1, (short)0, c, false, false);

            if (pv) {
                const int ibase = s * HW + p;
#pragma unroll
                for (int r = 0; r < 8; ++r)
                    pred[(r + 8 * half) * I_TOT + ibase] = c[r];
            }
        }
    }

    // init routing logits to zero
    for (int i = tid; i < I_TOT; i += 128) blog[i] = 0.0f;
    __syncthreads();

    // ---------------- Phase 2: dynamic routing (3 iterations) ---------------
    for (int r = 0; r < ROUTE_NUM; ++r) {
        // --- softmax over i: c = softmax(blog) ---
        float lmax = -3.402823466e38f;
        for (int i = tid; i < I_TOT; i += 128) lmax = fmaxf(lmax, blog[i]);
#pragma unroll
        for (int off = 16; off > 0; off >>= 1)
            lmax = fmaxf(lmax, __shfl_xor(lmax, off));
        if (lane == 0) red[wave] = lmax;
        __syncthreads();
        const float gmax = fmaxf(fmaxf(red[0], red[1]), fmaxf(red[2], red[3]));
        __syncthreads();

        float lsum = 0.0f;
        for (int i = tid; i < I_TOT; i += 128) {
            float e = __expf(blog[i] - gmax);
            cprob[i] = e;
            lsum += e;
        }
#pragma unroll
        for (int off = 16; off > 0; off >>= 1)
            lsum += __shfl_xor(lsum, off);
        if (lane == 0) red[wave] = lsum;
        __syncthreads();
        const float inv = 1.0f / (red[0] + red[1] + red[2] + red[3]);
        __syncthreads();

        // --- s[d] = sum_i pred[d][i] * c[i] ---
        float acc[OUT_DIM];
#pragma unroll
        for (int d = 0; d < OUT_DIM; ++d) acc[d] = 0.0f;
        for (int i = tid; i < I_TOT; i += 128) {
            const float ci = cprob[i] * inv;
#pragma unroll
            for (int d = 0; d < OUT_DIM; ++d)
                acc[d] += pred[d * I_TOT + i] * ci;
        }
#pragma unroll
        for (int d = 0; d < OUT_DIM; ++d) {
#pragma unroll
            for (int off = 16; off > 0; off >>= 1)
                acc[d] += __shfl_xor(acc[d], off);
        }
        if (lane == 0) {
#pragma unroll
            for (int d = 0; d < OUT_DIM; ++d) red[wave * 16 + d] = acc[d];
        }
        __syncthreads();
        if (tid < OUT_DIM)
            sv[tid] = red[tid] + red[16 + tid] + red[32 + tid] + red[48 + tid];
        __syncthreads();

        // --- squash (single lane) ---
        if (tid == 0) {
            float n2 = 0.0f;
#pragma unroll
            for (int d = 0; d < OUT_DIM; ++d) n2 += sv[d] * sv[d];
            const float n     = sqrtf(n2);
            const float scale = n2 / (1.0f + n2) / (n + EPS_SQ);
#pragma unroll
            for (int d = 0; d < OUT_DIM; ++d) vv[d] = sv[d] * scale;
        }
        __syncthreads();

        // --- logit update: blog[i] += sum_d v[d] * pred[d][i] ---
        if (r < ROUTE_NUM - 1) {
            for (int i = tid; i < I_TOT; i += 128) {
                float dot = 0.0f;
#pragma unroll
                for (int d = 0; d < OUT_DIM; ++d)
                    dot += vv[d] * pred[d * I_TOT + i];
                blog[i] += dot;
            }
            __syncthreads();
        }
    }

    if (tid < OUT_DIM)
        out[((size_t)bb * NUM_OUT + j) * OUT_DIM + tid] = vv[tid];
}

extern "C" void kernel_launch(void* const* d_in, const int* in_sizes, int n_in,
                              void* d_out, int out_size, void* d_ws, size_t ws_size,
                              hipStream_t stream) {
    const float* x     = (const float*)d_in[0];
    const float* w     = (const float*)d_in[1];
    const float* bconv = (const float*)d_in[2];
    float* out = (float*)d_out;

    const int bs = in_sizes[0] / (NUM_SHARED * IN_DIM * HW);  // 256
    dim3 grid(bs * NUM_OUT);   // one block per (batch, out-capsule)
    dim3 block(128);           // 4 waves (wave32)
    caps_route_kernel<<<grid, block, 0, stream>>>(x, w, bconv, out);
}